// PR_75239237091454
// MI455X (gfx1250) — compile-verified
//
#include <hip/hip_runtime.h>
#include <hip/hip_bf16.h>

// Problem constants (match reference)
#define NB   8
#define NT   4096
#define NCIN 32
#define NCOUT 32
#define NKP  8
#define NBT  (NB * NT)          // 32768
#define PI_F 3.14159265358979f
#define TWO_PI_F 6.28318530717959f

typedef float v2f __attribute__((ext_vector_type(2)));
typedef float v8f __attribute__((ext_vector_type(8)));

// ---------------------------------------------------------------------------
// Stage A: alpha[b,t,f] = sum_c x[b,t,c] * exp(-2*pi*i*c*f/32)
// Dense (32768 x 32) x (32 x 32) real matmul pair via V_WMMA_F32_16X16X4_F32.
// alpha stored transposed as [b][i][t] for coalesced downstream reads.
// ---------------------------------------------------------------------------
__global__ __launch_bounds__(256)
void PR_75239237091454_dft32(const float* __restrict__ x,
                             float* __restrict__ aRe, float* __restrict__ aIm) {
    const int tid  = threadIdx.x;
    const int lane = tid & 31;
    const int wv   = tid >> 5;
    const int rt   = blockIdx.x * 8 + wv;      // row tile 0..2047
    const int rowBase = rt * 16;
    const int mrow = lane & 15;
    const bool hi  = lane >= 16;
    const int f0   = lane & 15;                // column within tile

    v8f accRe0 = {}; v8f accIm0 = {};
    v8f accRe1 = {}; v8f accIm1 = {};

    #pragma unroll
    for (int k0 = 0; k0 < 32; k0 += 4) {
        const int c0 = k0 + (hi ? 2 : 0);
        // A fragment: rows of x (16x4 f32, 2 VGPRs)
        const int xidx = (rowBase + mrow) * NCIN + c0;
        v2f a; a.x = x[xidx]; a.y = x[xidx + 1];

        // B fragments: DFT matrix F[c][f], angle exactly reduced mod 32
        // tile 0: f = f0 ; tile 1: f = f0 + 16
        float s0a, c0a, s0b, c0b, s1a, c1a, s1b, c1b;
        const int fA = f0, fB = f0 + 16;
        __sincosf((PI_F / 16.0f) * (float)((c0 * fA) & 31), &s0a, &c0a);
        __sincosf((PI_F / 16.0f) * (float)(((c0 + 1) * fA) & 31), &s0b, &c0b);
        __sincosf((PI_F / 16.0f) * (float)((c0 * fB) & 31), &s1a, &c1a);
        __sincosf((PI_F / 16.0f) * (float)(((c0 + 1) * fB) & 31), &s1b, &c1b);

        v2f bRe0; bRe0.x = c0a;  bRe0.y = c0b;
        v2f bIm0; bIm0.x = -s0a; bIm0.y = -s0b;
        v2f bRe1; bRe1.x = c1a;  bRe1.y = c1b;
        v2f bIm1; bIm1.x = -s1a; bIm1.y = -s1b;

        accRe0 = __builtin_amdgcn_wmma_f32_16x16x4_f32(false, a, false, bRe0,
                                                       (short)0, accRe0, false, false);
        accIm0 = __builtin_amdgcn_wmma_f32_16x16x4_f32(false, a, false, bIm0,
                                                       (short)0, accIm0, false, false);
        accRe1 = __builtin_amdgcn_wmma_f32_16x16x4_f32(false, a, false, bRe1,
                                                       (short)0, accRe1, false, false);
        accIm1 = __builtin_amdgcn_wmma_f32_16x16x4_f32(false, a, false, bIm1,
                                                       (short)0, accIm1, false, false);
    }

    // Store C tiles into alpha[b][i][t]
    #pragma unroll
    for (int v = 0; v < 8; ++v) {
        const int m  = v + (hi ? 8 : 0);
        const int bt = rowBase + m;
        const int bb = bt >> 12;        // / NT
        const int t  = bt & (NT - 1);
        const int i0 = f0;
        const int i1 = f0 + 16;
        aRe[(bb * NCIN + i0) * NT + t] = accRe0[v];
        aIm[(bb * NCIN + i0) * NT + t] = accIm0[v];
        aRe[(bb * NCIN + i1) * NT + t] = accRe1[v];
        aIm[(bb * NCIN + i1) * NT + t] = accIm1[v];
    }
}

// ---------------------------------------------------------------------------
// Stage B: S[i][o][t] = sum_k res/(i*w_t - pole)   (complex)
// ---------------------------------------------------------------------------
__global__ __launch_bounds__(256)
void PR_75239237091454_ssum(const float* __restrict__ times,
                            const float* __restrict__ pre, const float* __restrict__ pim,
                            const float* __restrict__ rre, const float* __restrict__ rim,
                            float* __restrict__ SRe, float* __restrict__ SIm) {
    const int id = blockIdx.x * 256 + threadIdx.x;   // T*CIN*COUT threads
    const int t = id & (NT - 1);
    const int i = (id >> 12) & 31;
    const int o = id >> 17;
    const float dt = times[1] - times[0];
    const float base = TWO_PI_F / (dt * (float)NT);
    const float mf = (t < NT / 2) ? (float)t : (float)(t - NT);
    const float wt = base * mf;

    float sre = 0.0f, sim = 0.0f;
    const int pbase = (i * NCOUT + o) * NKP;
    #pragma unroll
    for (int k = 0; k < NKP; ++k) {
        const float dre = -pre[pbase + k];
        const float dim = wt - pim[pbase + k];
        const float inv = 1.0f / (dre * dre + dim * dim);
        const float rr = rre[pbase + k], ri = rim[pbase + k];
        sre += (rr * dre + ri * dim) * inv;
        sim += (ri * dre - rr * dim) * inv;
    }
    SRe[(i * NCOUT + o) * NT + t] = sre;
    SIm[(i * NCOUT + o) * NT + t] = sim;
}

// ---------------------------------------------------------------------------
// Stage C: lambda[b*32+o][t] = sum_i alpha[b][i][t] * S[i][o][t]
// ---------------------------------------------------------------------------
__global__ __launch_bounds__(256)
void PR_75239237091454_lambda(const float* __restrict__ aRe, const float* __restrict__ aIm,
                              const float* __restrict__ SRe, const float* __restrict__ SIm,
                              float* __restrict__ lamRe, float* __restrict__ lamIm) {
    const int id = blockIdx.x * 256 + threadIdx.x;   // B*COUT*T threads
    const int t = id & (NT - 1);
    const int o = (id >> 12) & 31;
    const int b = id >> 17;
    float lr = 0.0f, li = 0.0f;
    #pragma unroll 4
    for (int i = 0; i < NCIN; ++i) {
        const float ar = aRe[(b * NCIN + i) * NT + t];
        const float ai = aIm[(b * NCIN + i) * NT + t];
        const float sr = SRe[(i * NCOUT + o) * NT + t];
        const float si = SIm[(i * NCOUT + o) * NT + t];
        lr += ar * sr - ai * si;
        li += ar * si + ai * sr;
    }
    lamRe[(b * NCOUT + o) * NT + t] = lr;
    lamIm[(b * NCOUT + o) * NT + t] = li;
}

// ---------------------------------------------------------------------------
// Stage D: gamma[b][i][o][k] = -sum_t alpha[b][i][t] * K[t,i,o,k]
// One block per (i,o); wave w handles batch b=w; wave32 shuffle reduction.
// ---------------------------------------------------------------------------
__global__ __launch_bounds__(256)
void PR_75239237091454_gamma(const float* __restrict__ times,
                             const float* __restrict__ aRe, const float* __restrict__ aIm,
                             const float* __restrict__ pre, const float* __restrict__ pim,
                             const float* __restrict__ rre, const float* __restrict__ rim,
                             float* __restrict__ gRe, float* __restrict__ gIm) {
    const int i = blockIdx.x >> 5;
    const int o = blockIdx.x & 31;
    const int b = threadIdx.x >> 5;     // wave id == batch
    const int lane = threadIdx.x & 31;

    const float dt = times[1] - times[0];
    const float base = TWO_PI_F / (dt * (float)NT);

    float pr[NKP], pi_[NKP], rr[NKP], ri[NKP];
    const int pbase = (i * NCOUT + o) * NKP;
    #pragma unroll
    for (int k = 0; k < NKP; ++k) {
        pr[k] = pre[pbase + k]; pi_[k] = pim[pbase + k];
        rr[k] = rre[pbase + k]; ri[k] = rim[pbase + k];
    }

    float accR[NKP], accI[NKP];
    #pragma unroll
    for (int k = 0; k < NKP; ++k) { accR[k] = 0.0f; accI[k] = 0.0f; }

    const float* ar_p = aRe + (b * NCIN + i) * NT;
    const float* ai_p = aIm + (b * NCIN + i) * NT;

    for (int t = lane; t < NT; t += 32) {
        const float mf = (t < NT / 2) ? (float)t : (float)(t - NT);
        const float wt = base * mf;
        const float ar = ar_p[t];
        const float ai = ai_p[t];
        #pragma unroll
        for (int k = 0; k < NKP; ++k) {
            const float dre = -pr[k];
            const float dim = wt - pi_[k];
            const float inv = 1.0f / (dre * dre + dim * dim);
            const float kr = (rr[k] * dre + ri[k] * dim) * inv;
            const float ki = (ri[k] * dre - rr[k] * dim) * inv;
            accR[k] -= ar * kr - ai * ki;
            accI[k] -= ar * ki + ai * kr;
        }
    }

    #pragma unroll
    for (int k = 0; k < NKP; ++k) {
        #pragma unroll
        for (int off = 16; off > 0; off >>= 1) {
            accR[k] += __shfl_xor(accR[k], off, 32);
            accI[k] += __shfl_xor(accI[k], off, 32);
        }
    }
    if (lane == 0) {
        const int gb = ((b * NCIN + i) * NCOUT + o) * NKP;
        #pragma unroll
        for (int k = 0; k < NKP; ++k) { gRe[gb + k] = accR[k]; gIm[gb + k] = accI[k]; }
    }
}

// ---------------------------------------------------------------------------
// Stage E1: IFFT (radix-64 pass 1): z[r][n1][m0] =
//   (sum_m1 lam[r][m1*64+m0] * e^{+2*pi*i*m1*n1/64}) * e^{+2*pi*i*m0*n1/4096}
// ---------------------------------------------------------------------------
__global__ __launch_bounds__(256)
void PR_75239237091454_ifft1(const float* __restrict__ lamRe, const float* __restrict__ lamIm,
                             float* __restrict__ zRe, float* __restrict__ zIm) {
    const int id = blockIdx.x * 256 + threadIdx.x;   // 256*64*64 threads
    const int n1 = id & 63;
    const int m0 = (id >> 6) & 63;
    const int r  = id >> 12;

    float ws, wc;
    __sincosf((TWO_PI_F / 64.0f) * (float)n1, &ws, &wc);
    float prr = 1.0f, pri = 0.0f;
    float accR = 0.0f, accI = 0.0f;
    const float* lr_p = lamRe + r * NT + m0;
    const float* li_p = lamIm + r * NT + m0;
    #pragma unroll 8
    for (int m1 = 0; m1 < 64; ++m1) {
        const float lr = lr_p[m1 * 64];
        const float li = li_p[m1 * 64];
        accR += lr * prr - li * pri;
        accI += lr * pri + li * prr;
        const float nr = prr * wc - pri * ws;
        pri = prr * ws + pri * wc;
        prr = nr;
    }
    float ts, tc;
    __sincosf((TWO_PI_F / 4096.0f) * (float)(m0 * n1), &ts, &tc);
    const float oR = accR * tc - accI * ts;
    const float oI = accR * ts + accI * tc;
    zRe[(r * 64 + n1) * 64 + m0] = oR;
    zIm[(r * 64 + n1) * 64 + m0] = oI;
}

// ---------------------------------------------------------------------------
// Stage X2: out[r=b*32+o][t] = Re( sum_{i,k} gamma[b][i][o][k]*exp(pole*t) )
// gamma slice for this o staged in LDS (16 KB), shared across the 8 batches.
// (Writes out fully; ifft pass 2 then accumulates x1 on top.)
// ---------------------------------------------------------------------------
__global__ __launch_bounds__(256)
void PR_75239237091454_x2(const float* __restrict__ times,
                          const float* __restrict__ pre, const float* __restrict__ pim,
                          const float* __restrict__ gRe, const float* __restrict__ gIm,
                          float* __restrict__ out) {
    __shared__ float gr_s[NB * NCIN * NKP];   // [i][k][b]
    __shared__ float gi_s[NB * NCIN * NKP];
    const int o = blockIdx.x;
    const int t = blockIdx.y * 256 + threadIdx.x;

    for (int idx = threadIdx.x; idx < NB * NCIN * NKP; idx += 256) {
        const int b = idx & 7;
        const int ik = idx >> 3;
        const int i = ik >> 3;
        const int k = ik & 7;
        const int gidx = ((b * NCIN + i) * NCOUT + o) * NKP + k;
        gr_s[idx] = gRe[gidx];
        gi_s[idx] = gIm[gidx];
    }
    __syncthreads();

    const float tv = times[t];
    float acc[NB];
    #pragma unroll
    for (int b = 0; b < NB; ++b) acc[b] = 0.0f;

    for (int i = 0; i < NCIN; ++i) {
        #pragma unroll
        for (int k = 0; k < NKP; ++k) {
            const int pidx = (i * NCOUT + o) * NKP + k;
            const float er = __expf(pre[pidx] * tv);
            float sn, cs;
            __sincosf(pim[pidx] * tv, &sn, &cs);
            const float exr = er * cs;
            const float exi = er * sn;
            const int lbase = (i * NKP + k) * NB;
            #pragma unroll
            for (int b = 0; b < NB; ++b)
                acc[b] += gr_s[lbase + b] * exr - gi_s[lbase + b] * exi;
        }
    }
    #pragma unroll
    for (int b = 0; b < NB; ++b)
        out[(b * NCOUT + o) * NT + t] = acc[b];
}

// ---------------------------------------------------------------------------
// Stage E2: IFFT pass 2 (+1/T) accumulated into out:
//   out[r][n1+64*n0] += (1/4096) * Re( sum_m0 z[r][n1][m0]*e^{+2*pi*i*m0*n0/64} )
// ---------------------------------------------------------------------------
__global__ __launch_bounds__(256)
void PR_75239237091454_ifft2(const float* __restrict__ zRe, const float* __restrict__ zIm,
                             float* __restrict__ out) {
    const int id = blockIdx.x * 256 + threadIdx.x;   // 256*64*64 threads
    const int n0 = id & 63;
    const int n1 = (id >> 6) & 63;
    const int r  = id >> 12;

    float ws, wc;
    __sincosf((TWO_PI_F / 64.0f) * (float)n0, &ws, &wc);
    float prr = 1.0f, pri = 0.0f;
    float accR = 0.0f;
    const float* zr_p = zRe + (r * 64 + n1) * 64;
    const float* zi_p = zIm + (r * 64 + n1) * 64;
    #pragma unroll 8
    for (int m0 = 0; m0 < 64; ++m0) {
        accR += zr_p[m0] * prr - zi_p[m0] * pri;
        const float nr = prr * wc - pri * ws;
        pri = prr * ws + pri * wc;
        prr = nr;
    }
    const int t = n1 + 64 * n0;
    out[r * NT + t] += accR * (1.0f / (float)NT);
}

// ---------------------------------------------------------------------------
extern "C" void kernel_launch(void* const* d_in, const int* in_sizes, int n_in,
                              void* d_out, int out_size, void* d_ws, size_t ws_size,
                              hipStream_t stream) {
    const float* x     = (const float*)d_in[0];
    const float* times = (const float*)d_in[1];
    const float* p_re  = (const float*)d_in[2];
    const float* p_im  = (const float*)d_in[3];
    const float* r_re  = (const float*)d_in[4];
    const float* r_im  = (const float*)d_in[5];
    float* out = (float*)d_out;

    float* w = (float*)d_ws;
    float* aRe   = w;                     w += NB * NCIN * NT;        // 1M
    float* aIm   = w;                     w += NB * NCIN * NT;        // 1M
    float* SRe   = w;                     w += NCIN * NCOUT * NT;     // 4M
    float* SIm   = w;                     w += NCIN * NCOUT * NT;     // 4M
    float* lamRe = w;                     w += NB * NCOUT * NT;       // 1M
    float* lamIm = w;                     w += NB * NCOUT * NT;       // 1M
    float* gRe   = w;                     w += NB * NCIN * NCOUT * NKP; // 64K
    float* gIm   = w;                     w += NB * NCIN * NCOUT * NKP; // 64K
    float* zRe   = w;                     w += NB * NCOUT * NT;       // 1M
    float* zIm   = w;                     w += NB * NCOUT * NT;       // 1M

    // A: channel DFT via WMMA (2048 waves, 8 waves/block)
    PR_75239237091454_dft32<<<256, 256, 0, stream>>>(x, aRe, aIm);
    // B: S = sum_k residue/(i*w - pole)
    PR_75239237091454_ssum<<<(NT * NCIN * NCOUT) / 256, 256, 0, stream>>>(
        times, p_re, p_im, r_re, r_im, SRe, SIm);
    // C: lambda
    PR_75239237091454_lambda<<<(NB * NCOUT * NT) / 256, 256, 0, stream>>>(
        aRe, aIm, SRe, SIm, lamRe, lamIm);
    // D: gamma
    PR_75239237091454_gamma<<<NCIN * NCOUT, 256, 0, stream>>>(
        times, aRe, aIm, p_re, p_im, r_re, r_im, gRe, gIm);
    // E1: IFFT pass 1
    PR_75239237091454_ifft1<<<(256 * 64 * 64) / 256, 256, 0, stream>>>(
        lamRe, lamIm, zRe, zIm);
    // X2: writes out fully
    PR_75239237091454_x2<<<dim3(NCOUT, NT / 256), 256, 0, stream>>>(
        times, p_re, p_im, gRe, gIm, out);
    // E2: IFFT pass 2, accumulates x1 into out
    PR_75239237091454_ifft2<<<(256 * 64 * 64) / 256, 256, 0, stream>>>(zRe, zIm, out);
}